// AdaXbmTripletLoss_32298154066256
// MI455X (gfx1250) — compile-verified
//
#include <hip/hip_runtime.h>
#include <hip/hip_bf16.h>
#include <math.h>

typedef __attribute__((ext_vector_type(2))) float v2f;
typedef __attribute__((ext_vector_type(8))) float v8f;

#define BS      256
#define M_ROWS  32768
#define DIM     512
#define KNEG    10
#define MARGIN  0.1f
#define EPSV    1e-6f
#define TMARGIN 0.31622776601683794f   // sqrt(0.1)

// ---- workspace layout (bytes) ----
#define WS_NNEG 0                         // long long [256*10] = 20480
#define WS_NB   (20480)                   // float [32768]
#define WS_SB   (WS_NB + 131072)          // float [32768]
#define WS_NA   (WS_SB + 131072)          // float [256]
#define WS_SA   (WS_NA + 1024)            // float [256]
#define WS_PS   (WS_SA + 1024)            // float [256]  pos_sim
#define WS_DAP  (WS_PS + 1024)            // float [256]  d_ap
#define WS_HASQ (WS_DAP + 1024)           // int   [256]
#define WS_TOT  (WS_HASQ + 1024)          // float
#define WS_CNT  (WS_TOT + 4)              // unsigned

__device__ __forceinline__ float wave_sum_f32(float v) {
  #pragma unroll
  for (int m = 16; m >= 1; m >>= 1) v += __shfl_xor(v, m, 32);
  return v;
}
__device__ __forceinline__ int wave_sum_i32(int v) {
  #pragma unroll
  for (int m = 16; m >= 1; m >>= 1) v += __shfl_xor(v, m, 32);
  return v;
}

// One wave per "row job": norms/sums for inputs_row and q rows, plus
// pos_sim / d_ap per query.
__global__ void __launch_bounds__(256) stats_kernel(const float* __restrict__ col,
                                                    const float* __restrict__ row,
                                                    char* __restrict__ ws) {
  const int wid  = threadIdx.x >> 5;
  const int lane = threadIdx.x & 31;
  const int rid  = blockIdx.x * 8 + wid;

  float* nb  = (float*)(ws + WS_NB);
  float* sb  = (float*)(ws + WS_SB);
  float* na  = (float*)(ws + WS_NA);
  float* sa  = (float*)(ws + WS_SA);
  float* ps  = (float*)(ws + WS_PS);
  float* dap = (float*)(ws + WS_DAP);

  if (rid < M_ROWS) {                       // nb/sb over inputs_row
    const float* p = row + (long long)rid * DIM;
    float s = 0.f, s2 = 0.f;
    for (int k = lane; k < DIM; k += 32) { float x = p[k]; s += x; s2 += x * x; }
    s = wave_sum_f32(s); s2 = wave_sum_f32(s2);
    if (lane == 0) { sb[rid] = s; nb[rid] = s2; }
  } else if (rid < M_ROWS + BS) {           // na/sa over q rows
    const int i = rid - M_ROWS;
    const float* p = col + (long long)i * DIM;
    float s = 0.f, s2 = 0.f;
    for (int k = lane; k < DIM; k += 32) { float x = p[k]; s += x; s2 += x * x; }
    s = wave_sum_f32(s); s2 = wave_sum_f32(s2);
    if (lane == 0) { sa[i] = s; na[i] = s2; }
  } else if (rid < M_ROWS + 2 * BS) {       // pos_sim and d_ap
    const int i = rid - M_ROWS - BS;
    const float* q = col + (long long)i * DIM;
    const float* p = row + (long long)(BS + i) * DIM;
    float dt = 0.f, dd = 0.f;
    for (int k = lane; k < DIM; k += 32) {
      float a = q[k], b = p[k];
      dt += a * b;
      float df = a - b + EPSV;
      dd += df * df;
    }
    dt = wave_sum_f32(dt); dd = wave_sum_f32(dd);
    if (lane == 0) { ps[i] = dt; dap[i] = sqrtf(dd); }
  }
}

// Per query i: has_q, qloc (first match, else 0), gather nnegs[qloc]. Also
// zero the global accumulators (stream-ordered before the main kernel).
__global__ void match_kernel(const long long* __restrict__ targets_col,
                             const long long* __restrict__ qidxs,
                             const long long* __restrict__ nnegs,
                             char* __restrict__ ws) {
  const int i = threadIdx.x;               // one block of 256
  long long* mn = (long long*)(ws + WS_NNEG);
  int* hasq = (int*)(ws + WS_HASQ);
  const long long t = targets_col[i];
  int loc = 0, found = 0;
  for (int b = 0; b < BS; ++b) {
    if (!found && t == qidxs[b]) { loc = b; found = 1; }
  }
  hasq[i] = found;
  #pragma unroll
  for (int k = 0; k < KNEG; ++k) mn[i * KNEG + k] = nnegs[loc * KNEG + k];
  if (i == 0) { *(float*)(ws + WS_TOT) = 0.f; *(unsigned*)(ws + WS_CNT) = 0u; }
}

// Fused f32 WMMA GEMM (sim = q @ R^T) + triplet-loss epilogue.
// Wave -> 32(i) x 64(j) footprint = 2x4 accumulator tiles.
// Per K=4 step: 2 A-loads + 4 B-loads (b64) feed 8 v_wmma_f32_16x16x4_f32,
// 8 independent accumulation chains.
__global__ void __launch_bounds__(256) sim_loss_kernel(const float* __restrict__ col,
                                                       const float* __restrict__ row,
                                                       const long long* __restrict__ targets_row,
                                                       char* __restrict__ ws) {
  const int wid  = threadIdx.x >> 5;
  const int lane = threadIdx.x & 31;
  const int gw   = blockIdx.x * 8 + wid;   // 0..4095
  const int ip   = gw & 7;                 // 8 i-pairs (of 16-row tiles)
  const int jg   = gw >> 3;                // 512 j-groups of 64
  const int i0   = ip * 32;
  const int j0   = jg * 64;
  const int ln   = lane & 15;
  const int lg   = lane >> 4;              // half-wave: K pair / M half
  const int koff = lg * 2;                 // lanes 0-15: K=0,1 ; lanes 16-31: K=2,3

  const float* q0 = col + (long long)(i0 +  0 + ln) * DIM + koff;
  const float* q1 = col + (long long)(i0 + 16 + ln) * DIM + koff;
  const float* r0 = row + (long long)(j0 +  0 + ln) * DIM + koff;
  const float* r1 = row + (long long)(j0 + 16 + ln) * DIM + koff;
  const float* r2 = row + (long long)(j0 + 32 + ln) * DIM + koff;
  const float* r3 = row + (long long)(j0 + 48 + ln) * DIM + koff;

  v8f c00 = {}, c01 = {}, c02 = {}, c03 = {};
  v8f c10 = {}, c11 = {}, c12 = {}, c13 = {};
  #pragma unroll 2
  for (int kk = 0; kk < DIM; kk += 4) {
    v2f a0 = *(const v2f*)(q0 + kk);
    v2f a1 = *(const v2f*)(q1 + kk);
    v2f b0 = *(const v2f*)(r0 + kk);
    v2f b1 = *(const v2f*)(r1 + kk);
    v2f b2 = *(const v2f*)(r2 + kk);
    v2f b3 = *(const v2f*)(r3 + kk);
    c00 = __builtin_amdgcn_wmma_f32_16x16x4_f32(false, a0, false, b0, (short)0, c00, false, false);
    c01 = __builtin_amdgcn_wmma_f32_16x16x4_f32(false, a0, false, b1, (short)0, c01, false, false);
    c02 = __builtin_amdgcn_wmma_f32_16x16x4_f32(false, a0, false, b2, (short)0, c02, false, false);
    c03 = __builtin_amdgcn_wmma_f32_16x16x4_f32(false, a0, false, b3, (short)0, c03, false, false);
    c10 = __builtin_amdgcn_wmma_f32_16x16x4_f32(false, a1, false, b0, (short)0, c10, false, false);
    c11 = __builtin_amdgcn_wmma_f32_16x16x4_f32(false, a1, false, b1, (short)0, c11, false, false);
    c12 = __builtin_amdgcn_wmma_f32_16x16x4_f32(false, a1, false, b2, (short)0, c12, false, false);
    c13 = __builtin_amdgcn_wmma_f32_16x16x4_f32(false, a1, false, b3, (short)0, c13, false, false);
  }

  // ---- epilogue: consume sim tiles in registers ----
  const float*     nb   = (const float*)(ws + WS_NB);
  const float*     sb   = (const float*)(ws + WS_SB);
  const float*     na   = (const float*)(ws + WS_NA);
  const float*     sa   = (const float*)(ws + WS_SA);
  const float*     ps   = (const float*)(ws + WS_PS);
  const float*     dap  = (const float*)(ws + WS_DAP);
  const int*       hasq = (const int*)(ws + WS_HASQ);
  const long long* mn   = (const long long*)(ws + WS_NNEG);

  // j-side scalars: one set of 4 per lane, held in registers.
  long long tjv[4];
  float nbv[4], sbv[4];
  #pragma unroll
  for (int t = 0; t < 4; ++t) {
    const int j = j0 + t * 16 + ln;        // C/D layout: N = lane & 15
    tjv[t] = targets_row[j];
    nbv[t] = nb[j];
    sbv[t] = sb[j];
  }

  const float deps2 = (float)DIM * EPSV * EPSV;
  float total = 0.f;
  int   cnt   = 0;

  v8f cs[2][4] = { {c00, c01, c02, c03}, {c10, c11, c12, c13} };
  #pragma unroll
  for (int p = 0; p < 2; ++p) {
    #pragma unroll
    for (int r = 0; r < 8; ++r) {
      const int i = i0 + p * 16 + lg * 8 + r;  // C/D layout: VGPR r -> M = r (+8 hi lanes)
      // i-side scalars loaded once, reused across the 4 j-tiles.
      const float pos  = ps[i];
      const float nai  = na[i];
      const float sai  = sa[i];
      const float dapi = dap[i];
      const bool  hq   = (hasq[i] != 0);
      long long nn[KNEG];
      #pragma unroll
      for (int k = 0; k < KNEG; ++k) nn[k] = mn[i * KNEG + k];
      #pragma unroll
      for (int t = 0; t < 4; ++t) {
        const float s = cs[p][t][r];
        bool nonneg = false;
        #pragma unroll
        for (int k = 0; k < KNEG; ++k) nonneg = nonneg || (tjv[t] == nn[k]);
        const bool hard = (!nonneg) && (s > pos - MARGIN) && hq;
        const float dan2 = nai + nbv[t] - 2.f * s + 2.f * EPSV * (sai - sbv[t]) + deps2;
        const float dan  = sqrtf(fmaxf(dan2, 0.f));
        const float tl   = fmaxf(dapi - dan + TMARGIN, 0.f);
        if (hard) { total += tl; cnt += 1; }
      }
    }
  }

  total = wave_sum_f32(total);
  cnt   = wave_sum_i32(cnt);
  if (lane == 0) {
    atomicAdd((float*)(ws + WS_TOT), total);
    atomicAdd((unsigned*)(ws + WS_CNT), (unsigned)cnt);
  }
}

__global__ void finalize_kernel(const char* __restrict__ ws, float* __restrict__ out) {
  const float    t = *(const float*)(ws + WS_TOT);
  const unsigned c = *(const unsigned*)(ws + WS_CNT);
  out[0] = (c > 0u) ? (t / (float)c) : 0.f;
}

extern "C" void kernel_launch(void* const* d_in, const int* in_sizes, int n_in,
                              void* d_out, int out_size, void* d_ws, size_t ws_size,
                              hipStream_t stream) {
  const float*     col   = (const float*)d_in[0];       // inputs_col (512,512)
  const float*     row   = (const float*)d_in[1];       // inputs_row (32768,512)
  const long long* tcol  = (const long long*)d_in[2];   // targets_col int64
  const long long* trow  = (const long long*)d_in[3];   // targets_row int64
  const long long* qidxs = (const long long*)d_in[4];   // qidxs int64
  const long long* nnegs = (const long long*)d_in[6];   // nnegs (256,10) int64
  char* ws = (char*)d_ws;

  stats_kernel<<<(M_ROWS + 2 * BS) / 8, 256, 0, stream>>>(col, row, ws);
  match_kernel<<<1, 256, 0, stream>>>(tcol, qidxs, nnegs, ws);
  sim_loss_kernel<<<512, 256, 0, stream>>>(col, row, trow, ws);
  finalize_kernel<<<1, 1, 0, stream>>>(ws, (float*)d_out);
}